// MaxUpPool2DWithIndices_49581102465347
// MI455X (gfx1250) — compile-verified
//
#include <hip/hip_runtime.h>

// Max-unpooling (KSIZE=2) for B=32, H=128, W=128, C=64, f32.
// One thread owns one 2x2 output window x 4 channels:
//   - loads x[4] and indices[4] as b128 non-temporal
//   - decodes (dy,dx) per channel from the flat index (pure shifts: WO*C=2^14, C=2^6)
//   - writes all four output quadrants as b128 non-temporal stores
// Every output element is written exactly once -> no separate zero-fill pass,
// deterministic across graph replays (no atomics, no state carried over).

namespace {

constexpr int KS = 2;
constexpr int B = 32, H = 128, W = 128, C = 64;
constexpr int HO = H * KS;            // 256
constexpr int WO = W * KS;            // 256
constexpr int CQ = C / 4;             // 16 float4 groups per window
constexpr int THREADS = 256;
constexpr long long TOTAL_THREADS = (long long)B * H * W * CQ;  // 8,388,608

typedef float v4f __attribute__((ext_vector_type(4)));
typedef int   v4i __attribute__((ext_vector_type(4)));

__global__ __launch_bounds__(THREADS)
void max_unpool2d_scatter(const float* __restrict__ x,
                          const int*   __restrict__ ind,
                          float*       __restrict__ out)
{
    const unsigned tid = blockIdx.x * (unsigned)THREADS + threadIdx.x;

    // tid = ((b*H + h)*W + w)*CQ + cq   (all powers of two -> shifts/masks)
    const unsigned cq = tid & (CQ - 1);
    const unsigned w  = (tid >> 4)  & (W - 1);
    const unsigned h  = (tid >> 11) & (H - 1);
    const unsigned b  = tid >> 18;

    // Flat input offset of this channel-quad: exactly tid*4.
    const size_t in_off = (size_t)tid * 4u;

    // Streaming (non-temporal) b128 loads: no reuse, don't pollute L2.
    const v4f xv = __builtin_nontemporal_load((const v4f*)(x + in_off));
    const v4i iv = __builtin_nontemporal_load((const v4i*)(ind + in_off));

    // idx = (y*WO + xo)*C + c ; window guarantee: y in {2h,2h+1}, xo in {2w,2w+1}.
    // dy = (idx >> 14) & 1, dx = (idx >> 6) & 1 ; quadrant sel = 2*dy + dx.
    const int s0 = ((iv.x >> 13) & 2) | ((iv.x >> 6) & 1);
    const int s1 = ((iv.y >> 13) & 2) | ((iv.y >> 6) & 1);
    const int s2 = ((iv.z >> 13) & 2) | ((iv.z >> 6) & 1);
    const int s3 = ((iv.w >> 13) & 2) | ((iv.w >> 6) & 1);

    // Top-left of this window's 2x2 output block, at channel 4*cq.
    float* const obase =
        out + ((((size_t)b * HO + 2u * h) * WO + 2u * w) * (size_t)C + 4u * cq);

#pragma unroll
    for (int q = 0; q < 4; ++q) {
        v4f ov;
        ov.x = (s0 == q) ? xv.x : 0.0f;
        ov.y = (s1 == q) ? xv.y : 0.0f;
        ov.z = (s2 == q) ? xv.z : 0.0f;
        ov.w = (s3 == q) ? xv.w : 0.0f;
        float* const dst = obase + (size_t)(q >> 1) * ((size_t)WO * C)  // dy row step
                                 + (size_t)(q & 1) * (size_t)C;         // dx col step
        __builtin_nontemporal_store(ov, (v4f*)dst);
    }
}

}  // namespace

extern "C" void kernel_launch(void* const* d_in, const int* in_sizes, int n_in,
                              void* d_out, int out_size, void* d_ws, size_t ws_size,
                              hipStream_t stream)
{
    const float* x   = (const float*)d_in[0];
    const int*   ind = (const int*)d_in[1];
    float*       out = (float*)d_out;

    const int blocks = (int)(TOTAL_THREADS / THREADS);  // 32768, exact cover
    max_unpool2d_scatter<<<blocks, THREADS, 0, stream>>>(x, ind, out);

    (void)in_sizes; (void)n_in; (void)out_size; (void)d_ws; (void)ws_size;
}